// S4Layer_1090921693685
// MI455X (gfx1250) — compile-verified
//
#include <hip/hip_runtime.h>
#include <hip/hip_bf16.h>

// S4 diagonal SSM as chunked WMMA scan for gfx1250 (MI455X).
// y[b,t,d] = sum_{s<=t} K_d[t-s] x[b,s,d] + D_d x[b,t,d],
// computed chunk-wise with exact carried state so it matches the recurrence.
// Data staging uses the CDNA5 async global<->LDS path (ASYNCcnt).

#define DMODEL 256
#define NST    16
#define BSZ    16
#define SEQ    4096
#define TCH    16               // chunk length (== WMMA tile M)
#define SEGN   8                // parallel segments
#define SEGLEN (SEQ / SEGN)     // 512
#define CHUNKS (SEGLEN / TCH)   // 32
#define DPB    8                // channels per block (1 per wave32)

typedef __attribute__((ext_vector_type(2))) float v2f;
typedef __attribute__((ext_vector_type(8))) float v8f;

// D(16x16,f32) += A(16x16,f32) * B(16x16,f32) via 4x V_WMMA_F32_16X16X4_F32
static __device__ __forceinline__ v8f wmma_16x16x16_f32(const v2f a[4], const v2f b[4], v8f c) {
#pragma unroll
  for (int k = 0; k < 4; ++k)
    c = __builtin_amdgcn_wmma_f32_16x16x4_f32(false, a[k], false, b[k], (short)0, c,
                                              false, false);
  return c;
}

// Low 32 bits of a generic pointer to __shared__ == LDS byte address.
static __device__ __forceinline__ unsigned lds_addr(const void* p) {
  return (unsigned)(uintptr_t)p;
}

// Direct global -> LDS async copy (one DWORD), tracked by ASYNCcnt.
static __device__ __forceinline__ void async_load_dword(unsigned lds, const float* g) {
  asm volatile("global_load_async_to_lds_b32 %0, %1, off"
               :: "v"(lds), "v"((unsigned long long)(uintptr_t)g)
               : "memory");
}

// Direct LDS -> global async copy (one DWORD), tracked by ASYNCcnt.
static __device__ __forceinline__ void async_store_dword(float* g, unsigned lds) {
  asm volatile("global_store_async_from_lds_b32 %0, %1, off"
               :: "v"((unsigned long long)(uintptr_t)g), "v"(lds)
               : "memory");
}

static __device__ __forceinline__ void wait_async0() {
  asm volatile("s_wait_asynccnt 0x0" ::: "memory");
}

// ---------------------------------------------------------------------------
// Pass 0: per-channel derived matrices.
//   M[i][s] = (s<=i) ? K[i-s] : 0     (K[t] = sum_n C_n * Abar_n^t * Bbar_n)
//   W[i][n] = C_n * Abar_n^(i+1)
//   P[n][s] = Abar_n^(15-s) * Bbar_n
//   A16[n]  = Abar_n^16
// ---------------------------------------------------------------------------
__global__ __launch_bounds__(256)
void s4_prep(const float* __restrict__ A, const float* __restrict__ Bm,
             const float* __restrict__ Cm, const float* __restrict__ delta,
             float* __restrict__ wsM, float* __restrict__ wsP,
             float* __restrict__ wsW, float* __restrict__ wsA16) {
  const int d   = blockIdx.x;
  const int tid = threadIdx.x;
  __shared__ float Ab[NST], Bb[NST], Cv[NST], Kv[NST];
  __shared__ float Apow[TCH + 1][NST];

  const float sig = 1.0f / (1.0f + __expf(-delta[d]));
  if (tid < NST) {
    const float abar = __expf(sig * A[d * NST + tid]);
    Ab[tid] = abar;
    Bb[tid] = sig * Bm[d * NST + tid];
    Cv[tid] = Cm[d * NST + tid];
    float p = 1.0f;
    for (int k = 0; k <= TCH; ++k) { Apow[k][tid] = p; p *= abar; }
  }
  __syncthreads();
  if (tid < NST) {
    float s = 0.0f;
    for (int n = 0; n < NST; ++n) s += Cv[n] * Apow[tid][n] * Bb[n];
    Kv[tid] = s;
  }
  __syncthreads();
  const int r  = tid >> 4;  // row
  const int cc = tid & 15;  // col
  wsM[d * 256 + tid] = (cc <= r) ? Kv[r - cc] : 0.0f;       // M[i][s]
  wsW[d * 256 + tid] = Cv[cc] * Apow[r + 1][cc];            // W[i][n]
  wsP[d * 256 + tid] = Apow[15 - cc][r] * Bb[r];            // P[n][s]
  if (tid < NST) wsA16[d * NST + tid] = Apow[TCH][tid];
}

// ---------------------------------------------------------------------------
// Pass 1b: sequential combine of per-segment partial states (tiny).
//   Hinit[g] = state entering segment g;  Aseg = A16^32 = Abar^512
// ---------------------------------------------------------------------------
__global__ __launch_bounds__(256)
void s4_combine(const float* __restrict__ wsA16, const float* __restrict__ wsS,
                float* __restrict__ wsHinit) {
  const int d   = blockIdx.x;
  const int tid = threadIdx.x;        // n*16 + b
  const int n   = tid >> 4;
  float aseg = wsA16[d * NST + n];
  for (int k = 0; k < 5; ++k) aseg *= aseg;   // ^32
  float h = 0.0f;
  for (int g = 0; g < SEGN; ++g) {
    wsHinit[(d * SEGN + g) * 256 + tid] = h;
    h = aseg * h + wsS[(d * SEGN + g) * 256 + tid];
  }
}

// ---------------------------------------------------------------------------
// Main chunked scan. MODE 0: state-only pass (writes segment final states).
//                    MODE 1: full pass (writes y), starting from wsHinit.
// Block = 256 threads = 8 waves, wave w owns channel d = blockIdx.x*8 + w.
// ---------------------------------------------------------------------------
template <int MODE>
__global__ __launch_bounds__(256)
void s4_scan(const float* __restrict__ x, const float* __restrict__ Dskip,
             const float* __restrict__ wsM, const float* __restrict__ wsP,
             const float* __restrict__ wsW, const float* __restrict__ wsA16,
             const float* __restrict__ wsHinit, float* __restrict__ wsS,
             float* __restrict__ out) {
  const int dg   = blockIdx.x;        // channel group
  const int seg  = blockIdx.y;        // segment
  const int tid  = threadIdx.x;
  const int w    = tid >> 5;          // wave id == channel offset in group
  const int l    = tid & 31;          // lane
  const int d    = dg * DPB + w;
  const int b    = l & 15;            // N column (batch)
  const int half = l >> 4;            // lane half selects K/M sub-range

  __shared__ float Xs[DPB][TCH][BSZ];   // [d][s][b] input chunk
  __shared__ float Ys[DPB][TCH][BSZ];   // [d][i][b] output chunk
  __shared__ float Hs[DPB][NST][BSZ];   // [d][n][b] state relayout scratch

  // Per-wave constant operands (A-operand layout: lane = row, half/j = K).
  v2f Ma[4], Pa[4], Wa[4], A16v[4], Hb[4];
#pragma unroll
  for (int k = 0; k < 4; ++k) {
    const int kk = 4 * k + 2 * half;
    Ma[k][0] = wsM[d * 256 + b * 16 + kk];
    Ma[k][1] = wsM[d * 256 + b * 16 + kk + 1];
    Pa[k][0] = wsP[d * 256 + b * 16 + kk];
    Pa[k][1] = wsP[d * 256 + b * 16 + kk + 1];
    if (MODE == 1) {
      Wa[k][0] = wsW[d * 256 + b * 16 + kk];
      Wa[k][1] = wsW[d * 256 + b * 16 + kk + 1];
      Hb[k][0] = wsHinit[(d * SEGN + seg) * 256 + kk * 16 + b];
      Hb[k][1] = wsHinit[(d * SEGN + seg) * 256 + (kk + 1) * 16 + b];
    } else {
      Hb[k][0] = 0.0f;
      Hb[k][1] = 0.0f;
    }
    A16v[k][0] = wsA16[d * NST + kk];       // B-layout n index = kk+j
    A16v[k][1] = wsA16[d * NST + kk + 1];
  }
  const float dd = (MODE == 1) ? Dskip[d] : 0.0f;

  const int t0base = seg * SEGLEN;
  for (int c = 0; c < CHUNKS; ++c) {
    const int t0 = t0base + c * TCH;

    // Async global -> LDS staging of the X chunk for all 8 channels:
    // 8 consecutive threads fetch 8 consecutive channels (32B runs).
    {
      const int q    = tid & 7;
      const int base = tid >> 3;
#pragma unroll
      for (int it = 0; it < 8; ++it) {
        const int idx = base + it * 32;        // 0..255 -> (b,s)
        const int bb  = idx >> 4;
        const int ss  = idx & 15;
        async_load_dword(lds_addr(&Xs[q][ss][bb]),
                         x + ((size_t)bb * SEQ + t0 + ss) * DMODEL + dg * DPB + q);
      }
    }
    wait_async0();        // also retires last chunk's async Y stores
    __syncthreads();

    // X as B-operand (K = s dimension).
    v2f Xb[4];
#pragma unroll
    for (int k = 0; k < 4; ++k) {
      const int kk = 4 * k + 2 * half;
      Xb[k][0] = Xs[w][kk][b];
      Xb[k][1] = Xs[w][kk + 1][b];
    }

    // H_delta = P @ X  (D layout: rows n, cols b)
    v8f Hd = {0.f, 0.f, 0.f, 0.f, 0.f, 0.f, 0.f, 0.f};
    Hd = wmma_16x16x16_f32(Pa, Xb, Hd);

    if (MODE == 1) {
      // Y = W @ H_in + M @ X, then add skip D*x.
      v8f Y = {0.f, 0.f, 0.f, 0.f, 0.f, 0.f, 0.f, 0.f};
      Y = wmma_16x16x16_f32(Wa, Hb, Y);   // uses pre-update state
      Y = wmma_16x16x16_f32(Ma, Xb, Y);
#pragma unroll
      for (int v = 0; v < 8; ++v) {
        const int i = v + 8 * half;
        Ys[w][i][b] = Y[v] + dd * Xs[w][i][b];
      }
    }

    // Relayout H_delta (D layout) -> B layout through per-wave LDS region
    // (same-wave DS ops are ordered), then H = diag(A16)*H + H_delta.
#pragma unroll
    for (int v = 0; v < 8; ++v) Hs[w][v + 8 * half][b] = Hd[v];
#pragma unroll
    for (int k = 0; k < 4; ++k) {
      const int kk = 4 * k + 2 * half;
      Hb[k][0] = A16v[k][0] * Hb[k][0] + Hs[w][kk][b];
      Hb[k][1] = A16v[k][1] * Hb[k][1] + Hs[w][kk + 1][b];
    }
    __syncthreads();

    if (MODE == 1) {
      // Async LDS -> global writeback of the Y chunk (same 32B-run pattern).
      const int q    = tid & 7;
      const int base = tid >> 3;
#pragma unroll
      for (int it = 0; it < 8; ++it) {
        const int idx = base + it * 32;
        const int bb  = idx >> 4;
        const int ii  = idx & 15;
        async_store_dword(out + ((size_t)bb * SEQ + t0 + ii) * DMODEL + dg * DPB + q,
                          lds_addr(&Ys[q][ii][bb]));
      }
      // Retired by the wait at the next chunk's load (before Ys is rewritten)
      // or by S_ENDPGM's implicit wait-idle on the final chunk.
    }
  }

  if (MODE == 0) {
    // Segment-final state (zero-init) -> wsS[d][seg][n][b]
#pragma unroll
    for (int k = 0; k < 4; ++k) {
      const int kk = 4 * k + 2 * half;
      wsS[(d * SEGN + seg) * 256 + kk * 16 + b]       = Hb[k][0];
      wsS[(d * SEGN + seg) * 256 + (kk + 1) * 16 + b] = Hb[k][1];
    }
  }
}

// ---------------------------------------------------------------------------
extern "C" void kernel_launch(void* const* d_in, const int* in_sizes, int n_in,
                              void* d_out, int out_size, void* d_ws, size_t ws_size,
                              hipStream_t stream) {
  const float* x     = (const float*)d_in[0];
  const float* A     = (const float*)d_in[1];
  const float* Bm    = (const float*)d_in[2];
  const float* Cm    = (const float*)d_in[3];
  const float* Dsk   = (const float*)d_in[4];
  const float* delta = (const float*)d_in[5];
  float* out = (float*)d_out;

  float* wsf    = (float*)d_ws;
  float* wsM    = wsf;                    // 256*256
  float* wsP    = wsM + DMODEL * 256;     // 256*256
  float* wsW    = wsP + DMODEL * 256;     // 256*256
  float* wsA16  = wsW + DMODEL * 256;     // 256*16
  float* wsS    = wsA16 + DMODEL * NST;   // 256*8*256
  float* wsHini = wsS + DMODEL * SEGN * 256;  // 256*8*256  (~4.8 MB total)

  s4_prep<<<DMODEL, 256, 0, stream>>>(A, Bm, Cm, delta, wsM, wsP, wsW, wsA16);

  dim3 grid(DMODEL / DPB, SEGN);
  s4_scan<0><<<grid, 256, 0, stream>>>(x, Dsk, wsM, wsP, wsW, wsA16,
                                       nullptr, wsS, nullptr);

  s4_combine<<<DMODEL, 256, 0, stream>>>(wsA16, wsS, wsHini);

  s4_scan<1><<<grid, 256, 0, stream>>>(x, Dsk, wsM, wsP, wsW, wsA16,
                                       wsHini, nullptr, out);
}